// EdgeMaskNet_24129126269187
// MI455X (gfx1250) — compile-verified
//
#include <hip/hip_runtime.h>
#include <math.h>

#define HIDC 72
#define FINC 128
#define NPAD 80   // padded B width (5 x 16 column tiles)

typedef __attribute__((ext_vector_type(2))) float v2f;
typedef __attribute__((ext_vector_type(8))) float v8f;

__device__ __forceinline__ v8f wmma4(v2f a, v2f b, v8f c) {
  // D = A(16x4,f32) * B(4x16,f32) + C(16x16,f32)
  return __builtin_amdgcn_wmma_f32_16x16x4_f32(
      /*neg_a=*/false, a, /*neg_b=*/false, b,
      /*c_mod=*/(short)0, c, /*reuse_a=*/false, /*reuse_b=*/false);
}

// ---------------------------------------------------------------------------
// degree / norm
// ---------------------------------------------------------------------------
__global__ void k_deg(const int* __restrict__ colI, float* __restrict__ deg, int E) {
  int i = blockIdx.x * blockDim.x + threadIdx.x;
  if (i < E) atomicAdd(&deg[colI[i]], 1.0f);
}

__global__ void k_dinv(float* __restrict__ d, int N) {
  int i = blockIdx.x * blockDim.x + threadIdx.x;
  if (i < N) {
    float dv = d[i];
    d[i] = (dv > 0.0f) ? rsqrtf(fmaxf(dv, 1.0f)) : 0.0f;
  }
}

// ---------------------------------------------------------------------------
// Pad a K x 72 weight to K x 80 (zero-fill cols 72..79) so WMMA B-fragment
// loads need no column guards.
// ---------------------------------------------------------------------------
__global__ void k_pad_w(const float* __restrict__ src, float* __restrict__ dst,
                        int K) {
  const int idx = blockIdx.x * blockDim.x + threadIdx.x;
  if (idx >= K * NPAD) return;
  const int k = idx / NPAD;
  const int n = idx % NPAD;
  dst[idx] = (n < HIDC) ? src[k * HIDC + n] : 0.0f;
}

// ---------------------------------------------------------------------------
// WMMA-f32 GEMM: out[M x 72] = act(A[M x K] @ Bp[K x 80] (+bias)(+add))
// One wave computes 16 rows x 72 cols (5 column tiles).
// Requires: K % 4 == 0, M % 16 == 0. Bp is zero-padded to width 80.
// ---------------------------------------------------------------------------
template <bool HAS_BIAS, bool HAS_ADD, bool RELU>
__global__ void k_gemm72(const float* __restrict__ A, int lda, int K, int M,
                         const float* __restrict__ Bp,
                         const float* __restrict__ bias,
                         const float* __restrict__ add,
                         float* __restrict__ out) {
  const int lane = threadIdx.x & 31;
  const int wave = threadIdx.x >> 5;
  const int r    = lane & 15;
  const int half = lane >> 4;
  const int row0 = (blockIdx.x * (blockDim.x >> 5) + wave) * 16;
  if (row0 >= M) return;  // wave-uniform: EXEC stays all-ones for WMMA

  v8f acc[5] = {};
  const float* Arow = A + (size_t)(row0 + r) * lda;

  for (int k0 = 0; k0 < K; k0 += 4) {
    const int ka = k0 + 2 * half;
    v2f a;
    a.x = Arow[ka];          // contiguous pair -> global_load_b64
    a.y = Arow[ka + 1];
    const float* B0 = Bp + (size_t)ka * NPAD;
#pragma unroll
    for (int t = 0; t < 5; ++t) {
      v2f b;
      b.x = B0[t * 16 + r];
      b.y = B0[NPAD + t * 16 + r];
      acc[t] = wmma4(a, b, acc[t]);
    }
  }

#pragma unroll
  for (int t = 0; t < 5; ++t) {
    const int cc = t * 16 + r;
    if (cc >= HIDC) continue;  // only tile 4 partially masked
#pragma unroll
    for (int v = 0; v < 8; ++v) {
      const int rr = row0 + v + 8 * half;
      float val = acc[t][v];
      if (HAS_BIAS) val += bias[cc];
      if (HAS_ADD)  val += add[(size_t)rr * HIDC + cc];
      if (RELU)     val = fmaxf(val, 0.0f);
      out[(size_t)rr * HIDC + cc] = val;
    }
  }
}

// ---------------------------------------------------------------------------
// agg[col] += dinv[row]*dinv[col] * m[row]   (8 threads per edge, 9 ch each)
// ---------------------------------------------------------------------------
__global__ void k_scatter(const float* __restrict__ m, const int* __restrict__ rowI,
                          const int* __restrict__ colI, const float* __restrict__ dinv,
                          float* __restrict__ agg, int E) {
  const int e = blockIdx.x * 32 + (threadIdx.x >> 3);
  if (e >= E) return;
  const int rr = rowI[e];
  const int cc = colI[e];
  const float w = dinv[rr] * dinv[cc];
  const float* mr = m + (size_t)rr * HIDC;
  float* ag = agg + (size_t)cc * HIDC;
  const int c0 = (threadIdx.x & 7) * 9;
#pragma unroll
  for (int i = 0; i < 9; ++i) atomicAdd(&ag[c0 + i], w * mr[c0 + i]);
}

// ---------------------------------------------------------------------------
// BatchNorm (training statistics over N nodes)
// ---------------------------------------------------------------------------
__global__ void k_bn_reduce(const float* __restrict__ h, float* __restrict__ s1,
                            float* __restrict__ s2, int N) {
  // blockDim = 288: channel = tid%72, row-group = tid/72 (0..3); block covers 256 rows
  const int c = threadIdx.x % HIDC;
  const int g = threadIdx.x / HIDC;
  const int base = blockIdx.x * 256;
  float a = 0.0f, b = 0.0f;
  const int lim = (base + 256 < N) ? base + 256 : N;
  for (int r = base + g; r < lim; r += 4) {
    const float v = h[(size_t)r * HIDC + c];
    a += v;
    b += v * v;
  }
  atomicAdd(&s1[c], a);
  atomicAdd(&s2[c], b);
}

__global__ void k_bn_apply(float* __restrict__ h, const float* __restrict__ s1,
                           const float* __restrict__ s2, const float* __restrict__ gamma,
                           const float* __restrict__ beta, int N) {
  const int idx = blockIdx.x * blockDim.x + threadIdx.x;
  if (idx >= N * HIDC) return;
  const int c = idx % HIDC;
  const float invN = 1.0f / (float)N;
  const float mu = s1[c] * invN;
  const float var = s2[c] * invN - mu * mu;
  h[idx] = (h[idx] - mu) * rsqrtf(var + 1e-5f) * gamma[c] + beta[c];
}

// ---------------------------------------------------------------------------
// Pre-compose edge-head weights into padded Wcp[160 x 80]:
//   rows 0:72    = edge_lin1_w[0:72]   @ mlp_w1[0:72]      (h_row path)
//   rows 72:144  = edge_lin1_w[72:144] @ mlp_w1[0:72]      (h_col path)
//   rows 144:160 = edge_lin2_w         @ mlp_w1[72:144]    (edge_attr path)
//   cols 72..79  = 0
//   biasc        = b1 @ mlp_w1[0:72] + b2 @ mlp_w1[72:144] + mlp_b1
// ---------------------------------------------------------------------------
__global__ void k_fuse_w(const float* __restrict__ w1, const float* __restrict__ b1,
                         const float* __restrict__ w2, const float* __restrict__ b2,
                         const float* __restrict__ mw1, const float* __restrict__ mb1,
                         float* __restrict__ Wcp, float* __restrict__ biasc) {
  int idx = blockIdx.x * blockDim.x + threadIdx.x;
  for (; idx < 160 * NPAD; idx += gridDim.x * blockDim.x) {
    const int i = idx / NPAD;
    const int n = idx % NPAD;
    float s = 0.0f;
    if (n < HIDC) {
      if (i < 2 * HIDC) {
        for (int j = 0; j < HIDC; ++j) s += w1[i * HIDC + j] * mw1[j * HIDC + n];
      } else {
        const int i2 = i - 2 * HIDC;
        for (int j = 0; j < HIDC; ++j) s += w2[i2 * HIDC + j] * mw1[(HIDC + j) * HIDC + n];
      }
    }
    Wcp[idx] = s;
  }
  if (blockIdx.x == 0 && threadIdx.x < HIDC) {
    const int n = threadIdx.x;
    float s = mb1[n];
    for (int j = 0; j < HIDC; ++j)
      s += b1[j] * mw1[j * HIDC + n] + b2[j] * mw1[(HIDC + j) * HIDC + n];
    biasc[n] = s;
  }
}

// ---------------------------------------------------------------------------
// Fused edge head: per 16-edge tile
//   pre = [h[row] | h[col] | edge_attr] (16x160) @ Wcp (160x80) + biasc
//   out = tanh(pre) @ mlp_w2 + mlp_b2
// Feature tile staged in LDS via float4 gathers; requires E % 16 == 0.
// ---------------------------------------------------------------------------
#define EW 4          // waves per block
#define AK 160        // fused K
#define ASTR 164      // LDS row stride (floats): 164%4==0, conflict-free

__global__ void k_edge_head(const float* __restrict__ h, const int* __restrict__ eidx,
                            const float* __restrict__ ea, const float* __restrict__ Wcp,
                            const float* __restrict__ biasc, const float* __restrict__ w2,
                            const float* __restrict__ b2, float* __restrict__ out, int E) {
  __shared__ float sA[EW][16][ASTR];
  const int lane = threadIdx.x & 31;
  const int wave = threadIdx.x >> 5;
  const int r    = lane & 15;
  const int half = lane >> 4;
  const int e0   = (blockIdx.x * EW + wave) * 16;
  const int me   = e0 + r;                     // E % 16 == 0: always valid
  const int hr = eidx[me];
  const int hc = eidx[E + me];

  // Cooperative gather of the 16x160 feature tile (all float4-aligned:
  // h rows stride 288B from 256B-aligned base; ea rows 64B).
  {
    const float4* pr4 = (const float4*)(h + (size_t)hr * HIDC);
    const float4* pc4 = (const float4*)(h + (size_t)hc * HIDC);
    const float4* pe4 = (const float4*)(ea + (size_t)me * 16);
    float4* dst = (float4*)&sA[wave][r][0];
    if (half == 0) {              // k = 0..79:  h_row[0:72) + h_col[0:8)
#pragma unroll
      for (int i = 0; i < 18; ++i) dst[i] = pr4[i];
      dst[18] = pc4[0];
      dst[19] = pc4[1];
    } else {                      // k = 80..159: h_col[8:72) + ea[0:16)
#pragma unroll
      for (int i = 0; i < 16; ++i) dst[20 + i] = pc4[2 + i];
#pragma unroll
      for (int i = 0; i < 4; ++i) dst[36 + i] = pe4[i];
    }
  }
  __syncthreads();

  v8f acc[5] = {};
  const float* aRow = &sA[wave][r][0];
  for (int k0 = 0; k0 < AK; k0 += 4) {
    const int ka = k0 + 2 * half;
    v2f a;
    a.x = aRow[ka];              // contiguous, 8B-aligned -> ds_read_b64
    a.y = aRow[ka + 1];
    const float* B0 = Wcp + (size_t)ka * NPAD;
#pragma unroll
    for (int t = 0; t < 5; ++t) {
      v2f b;
      b.x = B0[t * 16 + r];
      b.y = B0[NPAD + t * 16 + r];
      acc[t] = wmma4(a, b, acc[t]);
    }
  }
  __syncthreads();               // done reading feature tile; reuse as out stage

  // tanh epilogue -> LDS (rows = edges in tile)
#pragma unroll
  for (int t = 0; t < 5; ++t) {
    const int cc = t * 16 + r;
    if (cc < HIDC) {
#pragma unroll
      for (int v = 0; v < 8; ++v)
        sA[wave][v + 8 * half][cc] = tanhf(acc[t][v] + biasc[cc]);
    }
  }
  __syncthreads();

  if (lane < 16) {
    float s = b2[0];
    for (int c = 0; c < HIDC; ++c) s += sA[wave][lane][c] * w2[c];
    out[e0 + lane] = s;
  }
}

// ---------------------------------------------------------------------------
// host launcher
// ---------------------------------------------------------------------------
extern "C" void kernel_launch(void* const* d_in, const int* in_sizes, int n_in,
                              void* d_out, int out_size, void* d_ws, size_t ws_size,
                              hipStream_t stream) {
  (void)n_in; (void)out_size; (void)ws_size;
  const float* x     = (const float*)d_in[0];
  const int*   eidx  = (const int*)d_in[1];
  const float* eattr = (const float*)d_in[2];
  const float* nlw   = (const float*)d_in[3];
  const float* nlb   = (const float*)d_in[4];
  const float* convW = (const float*)d_in[5];
  const float* convV = (const float*)d_in[6];
  const float* convB = (const float*)d_in[7];
  const float* bng   = (const float*)d_in[8];
  const float* bnb   = (const float*)d_in[9];
  const float* el1w  = (const float*)d_in[10];
  const float* el1b  = (const float*)d_in[11];
  const float* el2w  = (const float*)d_in[12];
  const float* el2b  = (const float*)d_in[13];
  const float* mw1   = (const float*)d_in[14];
  const float* mb1   = (const float*)d_in[15];
  const float* mw2   = (const float*)d_in[16];
  const float* mb2   = (const float*)d_in[17];

  const int N = in_sizes[0] / FINC;   // 50000 (multiple of 16)
  const int E = in_sizes[1] / 2;      // 800000 (multiple of 16)

  float* ws = (float*)d_ws;
  size_t off = 0;
  auto alloc = [&](size_t nf) {
    float* p = ws + off;
    off += (nf + 63) & ~(size_t)63;
    return p;
  };
  float* dnorm  = alloc((size_t)N);             // deg -> dinv in-place
  float* h      = alloc((size_t)N * HIDC);
  float* m      = alloc((size_t)N * HIDC);
  float* agg    = alloc((size_t)N * HIDC);
  float* bns    = alloc(2 * HIDC);              // [sum | sumsq]
  float* nlwP   = alloc((size_t)FINC * NPAD);   // padded node_lin_w
  float* convWP = alloc((size_t)3 * HIDC * NPAD);
  float* convVP = alloc((size_t)3 * HIDC * NPAD);
  float* Wcp    = alloc((size_t)AK * NPAD);     // fused padded edge weights
  float* bc     = alloc(HIDC);

  const int gemmGrid = (N + 63) / 64;           // 4 waves/block * 16 rows

  // pad all B matrices once per launch
  k_pad_w<<<(FINC * NPAD + 255) / 256, 256, 0, stream>>>(nlw, nlwP, FINC);
  for (int t = 0; t < 3; ++t) {
    k_pad_w<<<(HIDC * NPAD + 255) / 256, 256, 0, stream>>>(
        convW + (size_t)t * HIDC * HIDC, convWP + (size_t)t * HIDC * NPAD, HIDC);
    k_pad_w<<<(HIDC * NPAD + 255) / 256, 256, 0, stream>>>(
        convV + (size_t)t * HIDC * HIDC, convVP + (size_t)t * HIDC * NPAD, HIDC);
  }

  // degree / symmetric norm
  hipMemsetAsync(dnorm, 0, (size_t)N * sizeof(float), stream);
  k_deg<<<(E + 255) / 256, 256, 0, stream>>>(eidx + E, dnorm, E);
  k_dinv<<<(N + 255) / 256, 256, 0, stream>>>(dnorm, N);

  // h = relu(x @ node_lin_w + b)
  k_gemm72<true, false, true><<<gemmGrid, 128, 0, stream>>>(
      x, FINC, FINC, N, nlwP, nlb, nullptr, h);

  for (int t = 0; t < 3; ++t) {
    // m = h @ conv_w[t]
    k_gemm72<false, false, false><<<gemmGrid, 128, 0, stream>>>(
        h, HIDC, HIDC, N, convWP + (size_t)t * HIDC * NPAD, nullptr, nullptr, m);
    // agg = segment_sum(norm * m[row], col)
    hipMemsetAsync(agg, 0, (size_t)N * HIDC * sizeof(float), stream);
    k_scatter<<<(E + 31) / 32, 256, 0, stream>>>(m, eidx, eidx + E, dnorm, agg, E);
    // h = relu(agg + h @ conv_v[t] + conv_b[t])
    k_gemm72<true, true, true><<<gemmGrid, 128, 0, stream>>>(
        h, HIDC, HIDC, N, convVP + (size_t)t * HIDC * NPAD, convB + (size_t)t * HIDC,
        agg, h);
    // batchnorm (training stats)
    hipMemsetAsync(bns, 0, 2 * HIDC * sizeof(float), stream);
    k_bn_reduce<<<(N + 255) / 256, 288, 0, stream>>>(h, bns, bns + HIDC, N);
    k_bn_apply<<<((size_t)N * HIDC + 255) / 256, 256, 0, stream>>>(
        h, bns, bns + HIDC, bng + (size_t)t * HIDC, bnb + (size_t)t * HIDC, N);
  }

  // pre-compose edge-head weights, then fused edge head
  k_fuse_w<<<50, 256, 0, stream>>>(el1w, el1b, el2w, el2b, mw1, mb1, Wcp, bc);
  k_edge_head<<<E / (16 * EW), 32 * EW, 0, stream>>>(
      h, eidx, eattr, Wcp, bc, mw2, mb2, (float*)d_out, E);
}